// Encoder_25065429139538
// MI455X (gfx1250) — compile-verified
//
#include <hip/hip_runtime.h>
#include <hip/hip_bf16.h>
#include <stdint.h>

// ---------------------------------------------------------------------------
// GCN VGAE encoder for MI455X (gfx1250, wave32).
// Dense GEMMs use V_WMMA_F32_16X16X4_F32 (full fp32 precision).  Each wave
// owns a 16-row stripe and all NT/16 column tiles at once (acc array), so A
// fragments are loaded exactly once per K-step.  W is staged transposed in
// LDS with a +2 pad so each lane's B pair is one conflict-free ds_load_b64.
// Sparse aggregation is edge-parallel with global_atomic_add_f32; self-loops
// folded analytically into the finalize kernel.
// ---------------------------------------------------------------------------

typedef __attribute__((ext_vector_type(2))) float v2f;
typedef __attribute__((ext_vector_type(8))) float v8f;

// ------------------------------ zero fill ----------------------------------
__global__ void zero_kernel(float* __restrict__ p, long long n) {
    long long i = (long long)blockIdx.x * blockDim.x + threadIdx.x;
    if (i < n) p[i] = 0.0f;
}

// ------------------------- degree accumulation -----------------------------
__global__ void degree_kernel(const long long* __restrict__ dst,
                              const float* __restrict__ ew,
                              float* __restrict__ deg_w,
                              float* __restrict__ deg_1,
                              long long E) {
    long long e = (long long)blockIdx.x * blockDim.x + threadIdx.x;
    if (e >= E) return;
    long long d = dst[e];
    atomicAdd(&deg_w[d], ew[e]);
    atomicAdd(&deg_1[d], 1.0f);
}

// dinv = rsqrt(deg + 1)   (self-loop weight 1 => deg+1 > 0 always)
__global__ void rsqrt_kernel(float* __restrict__ deg, long long n) {
    long long i = (long long)blockIdx.x * blockDim.x + threadIdx.x;
    if (i < n) deg[i] = rsqrtf(deg[i] + 1.0f);
}

// ------------------------------ WMMA GEMM ----------------------------------
// C[M x NT] = A[M x K] * W[K x NT], row-major, fp32, NT in {32,64}, K in {64,128}.
// Block = 256 threads = 8 waves; each wave owns one 16-row tile and ALL NT/16
// column tiles (acc array), so each A fragment is loaded once per K-step.
// W is staged transposed (+2-float pad per row) in LDS: lane's B pair is one
// 8-byte-aligned, bank-conflict-free ds_load_b64.  M must be a multiple of 16.
template <int K, int NT>
__global__ void gemm_wmma_f32(const float* __restrict__ A,
                              const float* __restrict__ W,
                              float* __restrict__ C, int M) {
    constexpr int KP = K + 2;                 // padded LDS row stride (dwords)
    constexpr int NTILES = NT / 16;
    __shared__ float sWT[NT * KP];            // transposed: sWT[n*KP + k] = W[k*NT + n]
    for (int i = threadIdx.x; i < K * NT; i += blockDim.x) {
        int k = i / NT, n = i % NT;           // coalesced global read
        sWT[n * KP + k] = W[i];
    }
    __syncthreads();

    const int wave = threadIdx.x >> 5;
    const int lane = threadIdx.x & 31;
    const int rowTile = blockIdx.x * 8 + wave;          // 16 rows per tile
    if (rowTile * 16 >= M) return;                      // wave-uniform, EXEC stays all-1

    const int half = lane >> 4;               // 0: K 0,1 / M 0..7 ; 1: K 2,3 / M 8..15
    const int k0   = half * 2;
    const int mn   = lane & 15;               // A row within tile / B,C column

    const float* arow = A + (size_t)(rowTile * 16 + mn) * K + k0;

    v8f acc[NTILES];
#pragma unroll
    for (int nt = 0; nt < NTILES; ++nt) acc[nt] = (v8f){};

#pragma unroll
    for (int k = 0; k < K; k += 4) {
        v2f a = *(const v2f*)(arow + k);                       // global_load_b64
#pragma unroll
        for (int nt = 0; nt < NTILES; ++nt) {
            const int col = nt * 16 + mn;
            v2f b = *(const v2f*)(&sWT[col * KP + k + k0]);    // ds_load_b64
            acc[nt] = __builtin_amdgcn_wmma_f32_16x16x4_f32(
                false, a, false, b, (short)0, acc[nt], false, false);
        }
    }

    // C/D layout: VGPR r holds M=r (lanes 0-15) and M=r+8 (lanes 16-31), N=lane&15
#pragma unroll
    for (int nt = 0; nt < NTILES; ++nt) {
        const int col = nt * 16 + mn;
#pragma unroll
        for (int r = 0; r < 8; ++r) {
            int row = rowTile * 16 + r + 8 * half;
            C[(size_t)row * NT + col] = acc[nt][r];
        }
    }
}

// --------------------------- edge aggregation ------------------------------
// out[dst, f] += lin[src, f] * dinv[src] * w(e) * dinv[dst]
// One thread per (edge, feature); F = 1<<logF (32 or 64) so a wave shares an edge.
__global__ void aggregate_kernel(const float* __restrict__ lin,
                                 const float* __restrict__ dinv,
                                 const long long* __restrict__ src,
                                 const long long* __restrict__ dst,
                                 const float* __restrict__ ew,   // nullptr => weight 1
                                 float* __restrict__ out,
                                 long long E, int logF) {
    long long total = E << logF;
    long long idx = (long long)blockIdx.x * blockDim.x + threadIdx.x;
    if (idx >= total) return;
    long long e = idx >> logF;
    int f = (int)(idx & ((1LL << logF) - 1));
    long long s = src[e];
    long long d = dst[e];
    float w = ew ? ew[e] : 1.0f;
    float norm = dinv[s] * w * dinv[d];
    atomicAdd(&out[((size_t)d << logF) + f], lin[((size_t)s << logF) + f] * norm);
}

// ------------------------------ finalize -----------------------------------
// agg[i,f] += lin[i,f]*dinv[i]^2 (self-loop) + bias[f]; optional ReLU.
__global__ void finalize_kernel(float* __restrict__ agg,
                                const float* __restrict__ lin,
                                const float* __restrict__ dinv,
                                const float* __restrict__ bias,
                                long long n, int logF, int relu) {
    long long total = n << logF;
    long long idx = (long long)blockIdx.x * blockDim.x + threadIdx.x;
    if (idx >= total) return;
    long long node = idx >> logF;
    int f = (int)(idx & ((1LL << logF) - 1));
    float di = dinv[node];
    float v = agg[idx] + lin[idx] * di * di + bias[f];
    if (relu) v = fmaxf(v, 0.0f);
    agg[idx] = v;
}

// ---------------------------------------------------------------------------
extern "C" void kernel_launch(void* const* d_in, const int* in_sizes, int n_in,
                              void* d_out, int out_size, void* d_ws, size_t ws_size,
                              hipStream_t stream) {
    const float*      x    = (const float*)d_in[0];
    const long long*  eidx = (const long long*)d_in[1];   // int64 [2, E]
    const float*      ew   = (const float*)d_in[2];
    const float*      W1   = (const float*)d_in[3];
    const float*      b1   = (const float*)d_in[4];
    const float*      W2   = (const float*)d_in[5];
    const float*      b2   = (const float*)d_in[6];
    const float*      Wmu  = (const float*)d_in[7];
    const float*      bmu  = (const float*)d_in[8];
    const float*      Wls  = (const float*)d_in[9];
    const float*      bls  = (const float*)d_in[10];

    const long long N = in_sizes[0] / 128;   // 100000
    const long long E = in_sizes[2];         // 1600000
    const long long* src = eidx;
    const long long* dst = eidx + E;

    float* ws     = (float*)d_ws;
    float* deg_w  = ws;                 // N   -> becomes dinv_w
    float* deg_1  = ws + N;             // N   -> becomes dinv_1
    float* lin    = ws + 2 * N;         // 64N : lin1, then lin2, then mu|ls halves
    float* agg1   = ws + 66 * N;        // 64N
    float* agg2   = ws + 130 * N;       // 64N
    float* lin_mu = lin;                // 32N
    float* lin_ls = lin + 32 * N;       // 32N
    float* out_mu = (float*)d_out;      // 32N
    float* out_ls = (float*)d_out + 32 * N;

    const int B = 256;
    auto blocks = [](long long n) { return (unsigned)((n + 255) / 256); };

    // ---- zero accumulators ----
    zero_kernel<<<blocks(2 * N), B, 0, stream>>>(deg_w, 2 * N);         // deg_w + deg_1
    zero_kernel<<<blocks(64 * N), B, 0, stream>>>(agg1, 64 * N);
    zero_kernel<<<blocks(64 * N), B, 0, stream>>>(agg2, 64 * N);
    zero_kernel<<<blocks(64 * N), B, 0, stream>>>(out_mu, 64 * N);      // mu + logstd

    // ---- degrees -> dinv ----
    degree_kernel<<<blocks(E), B, 0, stream>>>(dst, ew, deg_w, deg_1, E);
    rsqrt_kernel<<<blocks(2 * N), B, 0, stream>>>(deg_w, 2 * N);        // in-place on both

    const unsigned gemmBlocks = (unsigned)((N / 16 + 7) / 8);

    // ---- layer 1: h = relu(gcn(x, W1, b1, ew)) ----
    gemm_wmma_f32<128, 64><<<gemmBlocks, B, 0, stream>>>(x, W1, lin, (int)N);
    aggregate_kernel<<<blocks(E << 6), B, 0, stream>>>(lin, deg_w, src, dst, ew, agg1, E, 6);
    finalize_kernel<<<blocks(N << 6), B, 0, stream>>>(agg1, lin, deg_w, b1, N, 6, 1);

    // ---- layer 2: h = gcn(h, W2, b2, ew) ----
    gemm_wmma_f32<64, 64><<<gemmBlocks, B, 0, stream>>>(agg1, W2, lin, (int)N);
    aggregate_kernel<<<blocks(E << 6), B, 0, stream>>>(lin, deg_w, src, dst, ew, agg2, E, 6);
    finalize_kernel<<<blocks(N << 6), B, 0, stream>>>(agg2, lin, deg_w, b2, N, 6, 0);

    // ---- heads: mu / logstd (unweighted norm) ----
    gemm_wmma_f32<64, 32><<<gemmBlocks, B, 0, stream>>>(agg2, Wmu, lin_mu, (int)N);
    gemm_wmma_f32<64, 32><<<gemmBlocks, B, 0, stream>>>(agg2, Wls, lin_ls, (int)N);
    aggregate_kernel<<<blocks(E << 5), B, 0, stream>>>(lin_mu, deg_1, src, dst, (const float*)nullptr, out_mu, E, 5);
    aggregate_kernel<<<blocks(E << 5), B, 0, stream>>>(lin_ls, deg_1, src, dst, (const float*)nullptr, out_ls, E, 5);
    finalize_kernel<<<blocks(N << 5), B, 0, stream>>>(out_mu, lin_mu, deg_1, bmu, N, 5, 0);
    finalize_kernel<<<blocks(N << 5), B, 0, stream>>>(out_ls, lin_ls, deg_1, bls, N, 5, 0);
}